// CNN_qulv_19533511262737
// MI455X (gfx1250) — compile-verified
//
#include <hip/hip_runtime.h>
#include <cstdint>

// ---------------------------------------------------------------------------
// Problem constants (fixed by the reference: x=(32,512,64,64) f32, k=256,
// curvature stencil w = [[-1,5,-1],[5,-16,5],[-1,5,-1]]/16).
// ---------------------------------------------------------------------------
#define HH 64
#define WW 64
#define TILE_ELEMS (HH * WW)   // 4096 floats = 16 KB
#define OH 62
#define OW 62
#define NOUT (OH * OW)         // 3844 VALID-conv outputs
#define NCAND 512
#define KSEL 256
#define NBATCH 32

typedef __attribute__((ext_vector_type(2))) float v2f;
typedef __attribute__((ext_vector_type(4))) float f32x4;
typedef __attribute__((ext_vector_type(8))) float v8f;

// Low 32 bits of a generic pointer into LDS == byte offset within the
// workgroup's LDS allocation (what the async-to-LDS VDST operand wants).
__device__ __forceinline__ uint32_t lds_offset_of(const void* p) {
  return (uint32_t)(uintptr_t)p;
}

// CDNA5 async global->LDS DMA, 16 bytes per lane, tracked with ASYNCcnt.
__device__ __forceinline__ void async_load_b128_to_lds(const void* gaddr,
                                                       uint32_t lds_off) {
  asm volatile("global_load_async_to_lds_b128 %0, %1, off"
               :: "v"(lds_off), "v"(gaddr)
               : "memory");
}

__device__ __forceinline__ void wait_asynccnt0() {
#if __has_builtin(__builtin_amdgcn_s_wait_asynccnt)
  __builtin_amdgcn_s_wait_asynccnt(0);
#else
  asm volatile("s_wait_asynccnt 0x0" ::: "memory");
#endif
}

// ---------------------------------------------------------------------------
// Kernel 1: per-image curvature score.
// One block = one 64x64 image. Async-stage tile in LDS, stencil from LDS,
// reduce 32 lane partials with one V_WMMA_F32_16X16X4_F32 (B = ones).
// ---------------------------------------------------------------------------
__global__ __launch_bounds__(256) void score_kernel(const float* __restrict__ x,
                                                    float* __restrict__ p) {
  __shared__ float tile[TILE_ELEMS];
  __shared__ float wsum[8];

  const int tid = threadIdx.x;
  const size_t base = (size_t)blockIdx.x * TILE_ELEMS;

  // 16 KB tile: 1024 B128 lane-transfers, 4 per thread, coalesced 512B/wave.
#pragma unroll
  for (int r = 0; r < 4; ++r) {
    const int chunk = tid + r * 256;                    // 16-byte chunk id
    async_load_b128_to_lds(x + base + (size_t)chunk * 4,
                           lds_offset_of(&tile[chunk * 4]));
  }
  wait_asynccnt0();     // this wave's DMA done
  __syncthreads();      // all waves' DMA visible in LDS

  // VALID 3x3 stencil: out = 5/16*(N+S+E+W) - 1/16*(diagonals) - center
  float acc = 0.0f;
  for (int o = tid; o < NOUT; o += 256) {
    const int i = o / OW + 1;
    const int j = (o - (i - 1) * OW) + 1;
    const float* r0 = &tile[(i - 1) * WW + j];
    const float* r1 = &tile[i * WW + j];
    const float* r2 = &tile[(i + 1) * WW + j];
    const float edge = r0[0] + r1[-1] + r1[1] + r2[0];
    const float corn = r0[-1] + r0[1] + r2[-1] + r2[1];
    acc += fabsf(0.3125f * edge - 0.0625f * corn - r1[0]);
  }

  // Wave-level reduction on the matrix pipe:
  // A(16x4): lanes 0-15 -> {K0=acc, K1=0}, lanes 16-31 -> {K2=acc, K3=0}
  // B(4x16) = ones (layout-independent)  =>  D[m][n] = acc[m] + acc[m+16].
  v2f a; a.x = acc;  a.y = 0.0f;
  v2f b; b.x = 1.0f; b.y = 1.0f;
  v8f c = {};
  c = __builtin_amdgcn_wmma_f32_16x16x4_f32(
      /*neg_a=*/false, a, /*neg_b=*/false, b,
      /*c_mod=*/(short)0, c, /*reuse_a=*/false, /*reuse_b=*/false);
  // lanes 0-15 hold D rows 0-7 (cols identical), lanes 16-31 rows 8-15:
  float s = c[0] + c[1] + c[2] + c[3] + c[4] + c[5] + c[6] + c[7];
  s += __shfl_xor(s, 16, 32);   // halves together -> full 32-lane sum

  const int wave = tid >> 5;
  if ((tid & 31) == 0) wsum[wave] = s;
  __syncthreads();
  if (tid == 0) {
    float t = 0.0f;
#pragma unroll
    for (int w = 0; w < 8; ++w) t += wsum[w];
    p[blockIdx.x] = t;
  }
}

// ---------------------------------------------------------------------------
// Kernel 2: per-batch top-256-of-512 by rank counting (matches lax.top_k:
// descending order, ties broken toward the lower index). Ranks are unique,
// so the compaction write is conflict-free and deterministic.
// ---------------------------------------------------------------------------
__global__ __launch_bounds__(256) void topk_kernel(const float* __restrict__ p,
                                                   int* __restrict__ idx) {
  __shared__ float s[NCAND];
  const int b = blockIdx.x;
  const int tid = threadIdx.x;
  s[tid]       = p[b * NCAND + tid];
  s[tid + 256] = p[b * NCAND + tid + 256];
  __syncthreads();

  for (int cnd = tid; cnd < NCAND; cnd += 256) {
    const float v = s[cnd];
    int rank = 0;
    for (int j = 0; j < NCAND; ++j) {
      const float u = s[j];
      rank += (u > v) || (u == v && j < cnd);
    }
    if (rank < KSEL) idx[b * KSEL + rank] = cnd;
  }
}

// ---------------------------------------------------------------------------
// Kernel 3: gather winners. One block per selected image; B128 copies with
// nontemporal stores (output is write-once, keep it out of L2).
// ---------------------------------------------------------------------------
__global__ __launch_bounds__(256) void gather_kernel(const float* __restrict__ x,
                                                     const int* __restrict__ idx,
                                                     float* __restrict__ out) {
  const int slot = blockIdx.x;          // b*KSEL + r
  const int b = slot >> 8;
  const int n = idx[slot];
  const f32x4* __restrict__ src =
      (const f32x4*)(x + ((size_t)(b * NCAND + n)) * TILE_ELEMS);
  f32x4* __restrict__ dst = (f32x4*)(out + (size_t)slot * TILE_ELEMS);
  const int tid = threadIdx.x;
#pragma unroll
  for (int r = 0; r < 4; ++r) {
    f32x4 v = src[tid + r * 256];
    __builtin_nontemporal_store(v, &dst[tid + r * 256]);
  }
}

// ---------------------------------------------------------------------------
// Host entry. d_in[0]=x (f32, 32*512*64*64), d_in[1]=weight (baked),
// d_in[2]=k (==256, baked). Workspace: 16384 f32 scores + 8192 i32 indices.
// ---------------------------------------------------------------------------
extern "C" void kernel_launch(void* const* d_in, const int* in_sizes, int n_in,
                              void* d_out, int out_size, void* d_ws, size_t ws_size,
                              hipStream_t stream) {
  (void)in_sizes; (void)n_in; (void)out_size; (void)ws_size;
  const float* x = (const float*)d_in[0];
  float* out = (float*)d_out;

  float* p  = (float*)d_ws;                                        // 64 KB
  int*  idx = (int*)((char*)d_ws + NBATCH * NCAND * sizeof(float)); // 32 KB

  score_kernel <<<NBATCH * NCAND, 256, 0, stream>>>(x, p);
  topk_kernel  <<<NBATCH,         256, 0, stream>>>(p, idx);
  gather_kernel<<<NBATCH * KSEL,  256, 0, stream>>>(x, idx, out);
}